// BlockSparseSelfAttention_3427383902308
// MI455X (gfx1250) — compile-verified
//
#include <hip/hip_runtime.h>
#include <math.h>

typedef __attribute__((ext_vector_type(2))) float v2f;
typedef __attribute__((ext_vector_type(8))) float v8f;

#define B_   2
#define H_   16
#define S_   2048
#define D_   64
#define BS_  64
#define SCALE_ 0.125f   // 1/sqrt(64)

// ---------------------------------------------------------------------------
// Kernel 1: scores = mask(QK^T * scale)  -> attn region (raw, -inf at masked)
// One wave (32 threads) computes a 16x64 score tile via 4x16 WMMA f32 16x16x4.
// ---------------------------------------------------------------------------
__global__ __launch_bounds__(32)
void qk_scores_kernel(const float* __restrict__ Q,
                      const float* __restrict__ K,
                      float* __restrict__ attn) {
    const int bh    = blockIdx.z;
    const int row0  = blockIdx.y * 16;
    const int col0  = blockIdx.x * 64;
    const int lane  = threadIdx.x;
    const int m     = lane & 15;        // M (A) / N (B,C,D) index
    const int khalf = lane >> 4;        // selects K pair {0,1} vs {2,3}

    const float* Qb = Q    + (size_t)bh * S_ * D_;
    const float* Kb = K    + (size_t)bh * S_ * D_;
    float*       Ab = attn + (size_t)bh * S_ * S_;

    // Preload all A fragments for this 16-row strip (D=64 -> 16 k-steps of 4)
    v2f a[16];
#pragma unroll
    for (int s = 0; s < 16; ++s)
        a[s] = *(const v2f*)(Qb + (size_t)(row0 + m) * D_ + 4 * s + 2 * khalf);

#pragma unroll
    for (int nt = 0; nt < 4; ++nt) {
        const int col = col0 + nt * 16;
        v8f acc = {};
#pragma unroll
        for (int s = 0; s < 16; ++s) {
            // B[k][n] = K[col+n][k]: contiguous float2 along K's head dim
            v2f b = *(const v2f*)(Kb + (size_t)(col + m) * D_ + 4 * s + 2 * khalf);
            acc = __builtin_amdgcn_wmma_f32_16x16x4_f32(
                false, a[s], false, b, (short)0, acc, false, false);
        }
        // C/D layout: VGPR r, lanes 0-15 -> M=r, lanes 16-31 -> M=r+8; N=lane&15
#pragma unroll
        for (int r = 0; r < 8; ++r) {
            const int i = row0 + r + 8 * khalf;
            const int j = col + m;
            const bool masked = ((i >> 6) == (j >> 6)) || ((j & (BS_ - 1)) == 0);
            Ab[(size_t)i * S_ + j] = masked ? -__builtin_inff() : acc[r] * SCALE_;
        }
    }
}

// ---------------------------------------------------------------------------
// Kernel 2: row-wise softmax over attn (in place). One wave per row.
// ---------------------------------------------------------------------------
__global__ __launch_bounds__(256)
void softmax_kernel(float* __restrict__ attn) {
    const int lane = threadIdx.x & 31;
    const int wave = threadIdx.x >> 5;
    const size_t row = (size_t)blockIdx.x * 8 + wave;   // 8 waves per block
    float* base = attn + row * S_;

    float ev[S_ / 32];
    float mx = -__builtin_inff();
#pragma unroll
    for (int k = 0; k < S_ / 32; ++k) {
        ev[k] = base[lane + 32 * k];
        mx = fmaxf(mx, ev[k]);
    }
#pragma unroll
    for (int off = 16; off > 0; off >>= 1)
        mx = fmaxf(mx, __shfl_xor(mx, off, 32));

    float sum = 0.0f;
#pragma unroll
    for (int k = 0; k < S_ / 32; ++k) {
        ev[k] = __expf(ev[k] - mx);                     // exp(-inf)=0 at masked
        sum += ev[k];
    }
#pragma unroll
    for (int off = 16; off > 0; off >>= 1)
        sum += __shfl_xor(sum, off, 32);

    const float inv = 1.0f / sum;
#pragma unroll
    for (int k = 0; k < S_ / 32; ++k)
        base[lane + 32 * k] = ev[k] * inv;
}

// ---------------------------------------------------------------------------
// Kernel 3: out = attn @ V. One wave per 16x64 output strip; fp32 WMMA,
// accumulate over S in K-steps of 4.
// ---------------------------------------------------------------------------
__global__ __launch_bounds__(32)
void av_kernel(const float* __restrict__ V,
               const float* __restrict__ attn,
               float* __restrict__ out) {
    const int bh    = blockIdx.y;
    const int row0  = blockIdx.x * 16;
    const int lane  = threadIdx.x;
    const int m     = lane & 15;
    const int khalf = lane >> 4;

    const float* Vb = V    + (size_t)bh * S_ * D_;
    const float* Ab = attn + (size_t)bh * S_ * S_;
    float*       Ob = out  + (size_t)bh * S_ * D_;

    v8f acc[4] = {{}, {}, {}, {}};

#pragma unroll 4
    for (int t0 = 0; t0 < S_; t0 += 4) {
        // A fragment: attn rows, contiguous float2 in K (= t) dimension
        v2f a = *(const v2f*)(Ab + (size_t)(row0 + m) * S_ + t0 + 2 * khalf);
        // B fragment base: V[t0+2*khalf][.. + n], second K element one row down
        const float* vp = Vb + (size_t)(t0 + 2 * khalf) * D_ + m;
#pragma unroll
        for (int dt = 0; dt < 4; ++dt) {
            v2f b;
            b.x = vp[dt * 16];
            b.y = vp[dt * 16 + D_];
            acc[dt] = __builtin_amdgcn_wmma_f32_16x16x4_f32(
                false, a, false, b, (short)0, acc[dt], false, false);
        }
    }

#pragma unroll
    for (int dt = 0; dt < 4; ++dt)
#pragma unroll
        for (int r = 0; r < 8; ++r)
            Ob[(size_t)(row0 + r + 8 * khalf) * D_ + dt * 16 + m] = acc[dt][r];
}

// ---------------------------------------------------------------------------
// Kernel 4: mask M as float 0/1 (third tuple output)
// ---------------------------------------------------------------------------
__global__ __launch_bounds__(256)
void mask_kernel(float* __restrict__ Mout) {
    const size_t idx = (size_t)blockIdx.x * 256 + threadIdx.x;
    const int i = (int)(idx / S_);
    const int j = (int)(idx % S_);
    const bool mk = ((i >> 6) == (j >> 6)) || ((j & (BS_ - 1)) == 0);
    Mout[idx] = mk ? 1.0f : 0.0f;
}

// ---------------------------------------------------------------------------
extern "C" void kernel_launch(void* const* d_in, const int* in_sizes, int n_in,
                              void* d_out, int out_size, void* d_ws, size_t ws_size,
                              hipStream_t stream) {
    const float* Q = (const float*)d_in[0];
    const float* K = (const float*)d_in[1];
    const float* V = (const float*)d_in[2];

    float* out  = (float*)d_out;                          // B*H*S*D
    float* attn = out  + (size_t)B_ * H_ * S_ * D_;       // B*H*S*S
    float* Mf   = attn + (size_t)B_ * H_ * S_ * S_;       // S*S

    // 1) masked scaled scores into attn region
    qk_scores_kernel<<<dim3(S_ / 64, S_ / 16, B_ * H_), 32, 0, stream>>>(Q, K, attn);
    // 2) softmax in place
    softmax_kernel<<<(B_ * H_ * S_) / 8, 256, 0, stream>>>(attn);
    // 3) out = attn @ V
    av_kernel<<<dim3(S_ / 16, B_ * H_), 32, 0, stream>>>(V, attn, out);
    // 4) mask output
    mask_kernel<<<(S_ * (size_t)S_) / 256, 256, 0, stream>>>(Mf);
}